// MatchingLayer_21715354648692
// MI455X (gfx1250) — compile-verified
//
#include <hip/hip_runtime.h>

typedef __attribute__((ext_vector_type(2))) float v2f;
typedef __attribute__((ext_vector_type(4))) float v4f;
typedef __attribute__((ext_vector_type(8))) float v8f;

#define BATCH 16
#define LDIM  1024
#define DDIM  1024
#define TILE  128          // C tile per workgroup (128x128)
#define KC    32           // K chunk staged in LDS
#define LDST  36           // LDS row stride in floats (KC + 4 pad)

// ---------------------------------------------------------------------------
// Phase 1: per-row inverse norms of (x * w).  One 256-thread block per row.
// rows [0, B*L) -> a ; rows [B*L, 2*B*L) -> b
// ---------------------------------------------------------------------------
__global__ void __launch_bounds__(256)
matching_rownorm_kernel(const float* __restrict__ a,
                        const float* __restrict__ b,
                        const float* __restrict__ w,
                        float* __restrict__ inv_norms) {
    __shared__ float red[256];
    int row = blockIdx.x;                      // 0 .. 2*B*L-1
    const float* src = (row < BATCH * LDIM) ? a : b;
    int r = (row < BATCH * LDIM) ? row : row - BATCH * LDIM;
    const float* p = src + (size_t)r * DDIM;
    int tid = threadIdx.x;

    float s = 0.f;
    for (int i = tid; i < DDIM; i += 256) {
        float x = p[i] * w[i];
        s += x * x;
    }
    red[tid] = s;
    __syncthreads();
    #pragma unroll
    for (int off = 128; off > 0; off >>= 1) {
        if (tid < off) red[tid] += red[tid + off];
        __syncthreads();
    }
    if (tid == 0) inv_norms[row] = rsqrtf(fmaxf(red[0], 1e-12f));
}

// ---------------------------------------------------------------------------
// Phase 2: out[b,i,j] = inv_na[i]*inv_nb[j] * sum_d (a[b,i,d]*w[d])*(b[b,j,d]*w[d])
// 128x128 C tile per 256-thread workgroup (8 waves, each wave: 32x64 = 2x4
// subtiles of 16x16 computed with V_WMMA_F32_16X16X4_F32).
// ---------------------------------------------------------------------------
__global__ void __launch_bounds__(256)
matching_gemm_kernel(const float* __restrict__ A,
                     const float* __restrict__ Bm,
                     const float* __restrict__ w,
                     const float* __restrict__ inv,
                     float* __restrict__ out) {
    __shared__ float As[TILE * LDST];
    __shared__ float Bs[TILE * LDST];

    const int batch = blockIdx.z;
    const int tileM = blockIdx.x * TILE;
    const int tileN = blockIdx.y * TILE;
    const int tid   = threadIdx.x;
    const int lane  = tid & 31;
    const int wv    = tid >> 5;
    const int waveM = (wv & 3) * 32;   // 4 waves along M
    const int waveN = (wv >> 2) * 64;  // 2 waves along N

    const float* Ab = A  + (size_t)batch * LDIM * DDIM;
    const float* Bb = Bm + (size_t)batch * LDIM * DDIM;

    v8f acc[2][4];
    #pragma unroll
    for (int mi = 0; mi < 2; ++mi)
        #pragma unroll
        for (int ni = 0; ni < 4; ++ni) {
            v8f z = {};
            acc[mi][ni] = z;
        }

    // fp32 WMMA A layout: lane (l%16) = M row; lanes 0-15 hold K=k0,k0+1 and
    // lanes 16-31 hold K=k0+2,k0+3 as a float2.  B (stored [n][k]) mirrors it.
    const int mrow = lane & 15;
    const int koff = (lane < 16) ? 0 : 2;

    for (int k0 = 0; k0 < DDIM; k0 += KC) {
        // ---- stage 128xKC panels of A and B into LDS, scaled by w ----
        #pragma unroll
        for (int t = 0; t < 4; ++t) {
            int idx = tid + t * 256;          // 0..1023
            int row = idx >> 3;               // 8 float4 per KC row
            int col = (idx & 7) << 2;
            v4f wa = *(const v4f*)(w + k0 + col);
            v4f av = *(const v4f*)(Ab + (size_t)(tileM + row) * DDIM + k0 + col);
            v4f bv = *(const v4f*)(Bb + (size_t)(tileN + row) * DDIM + k0 + col);
            av *= wa;
            bv *= wa;
            *(v4f*)(&As[row * LDST + col]) = av;
            *(v4f*)(&Bs[row * LDST + col]) = bv;
        }
        __syncthreads();

        // ---- 8 k-steps of 4, 8 WMMAs per step per wave ----
        #pragma unroll
        for (int kk = 0; kk < KC; kk += 4) {
            v2f aF[2], bF[4];
            #pragma unroll
            for (int mi = 0; mi < 2; ++mi)
                aF[mi] = *(const v2f*)(&As[(waveM + mi * 16 + mrow) * LDST + kk + koff]);
            #pragma unroll
            for (int ni = 0; ni < 4; ++ni)
                bF[ni] = *(const v2f*)(&Bs[(waveN + ni * 16 + mrow) * LDST + kk + koff]);
            #pragma unroll
            for (int mi = 0; mi < 2; ++mi)
                #pragma unroll
                for (int ni = 0; ni < 4; ++ni)
                    acc[mi][ni] = __builtin_amdgcn_wmma_f32_16x16x4_f32(
                        false, aF[mi], false, bF[ni],
                        (short)0, acc[mi][ni], false, false);
        }
        __syncthreads();
    }

    // ---- apply inverse norms and store ----
    const float* invA = inv;                       // [B*L]
    const float* invB = inv + BATCH * LDIM;        // [B*L]
    const int mBase = tileM + waveM;
    const int nBase = tileN + waveN;
    const int rowHalf = (lane >> 4) << 3;          // 0 or 8
    const int ncol = lane & 15;

    #pragma unroll
    for (int mi = 0; mi < 2; ++mi) {
        #pragma unroll
        for (int r = 0; r < 8; ++r) {
            int row = mBase + mi * 16 + r + rowHalf;
            float ia = invA[batch * LDIM + row];
            #pragma unroll
            for (int ni = 0; ni < 4; ++ni) {
                int col = nBase + ni * 16 + ncol;
                float ib = invB[batch * LDIM + col];
                out[((size_t)batch * LDIM + row) * LDIM + col] =
                    acc[mi][ni][r] * ia * ib;
            }
        }
    }
}

// ---------------------------------------------------------------------------
extern "C" void kernel_launch(void* const* d_in, const int* in_sizes, int n_in,
                              void* d_out, int out_size, void* d_ws, size_t ws_size,
                              hipStream_t stream) {
    const float* a = (const float*)d_in[0];
    const float* b = (const float*)d_in[1];
    const float* w = (const float*)d_in[2];   // kernel[D,1] -> D floats
    float* inv = (float*)d_ws;                // 2*B*L floats = 128 KB
    float* out = (float*)d_out;

    matching_rownorm_kernel<<<2 * BATCH * LDIM, 256, 0, stream>>>(a, b, w, inv);

    dim3 grid(LDIM / TILE, LDIM / TILE, BATCH);
    matching_gemm_kernel<<<grid, 256, 0, stream>>>(a, b, w, inv, out);
}